// CP_Based_4183298146463
// MI455X (gfx1250) — compile-verified
//
#include <hip/hip_runtime.h>

typedef __attribute__((ext_vector_type(2))) float v2f;
typedef __attribute__((ext_vector_type(8))) float v8f;

#define NB 1024   // batch
#define NF 64     // num_feats
#define ND 4      // local_dim (== WMMA K)
#define NR 32     // cp_rank
#define NU 128    // units

// Block: 256 threads = 8 wave32s. Grid: (NB/16) x (NU/16) = 64 x 8.
// Each wave handles 4 values of r; per feature f it issues 4 independent
// V_WMMA_F32_16X16X4_F32 ops and folds each 16x16 result into a running
// elementwise product (Hadamard over features).
__global__ __launch_bounds__(256) void cp_wmma_kernel(const float* __restrict__ X,
                                                      const float* __restrict__ W,
                                                      float* __restrict__ out) {
    const int btile = blockIdx.x;        // 0..63
    const int utile = blockIdx.y;        // 0..7
    const int tid   = threadIdx.x;
    const int lane  = tid & 31;
    const int wave  = tid >> 5;          // 0..7

    const int b0 = btile * 16;
    const int u0 = utile * 16;

    const int half = lane >> 4;          // 0 = lanes 0-15, 1 = lanes 16-31
    const int l16  = lane & 15;

    // A-matrix lane addressing (16x4 f32): lane row m = l16,
    // K pair = {2*half, 2*half+1}. X[b, f, d] with d contiguous -> b64 load.
    const float* xa = X + (size_t)(b0 + l16) * (NF * ND) + (size_t)(half * 2);

    // B-matrix lane addressing (4x16 f32): n = l16, K rows {2*half, 2*half+1}.
    // W[d, r, f, u] index = d*(NR*NF*NU) + r*(NF*NU) + f*NU + u.
    const size_t dplane = (size_t)NR * NF * NU;      // stride between d planes
    const float* wb = W + (size_t)(half * 2) * dplane + (size_t)u0 + (size_t)l16;

    const v8f vzero = {0.f, 0.f, 0.f, 0.f, 0.f, 0.f, 0.f, 0.f};
    v8f prod[4];
    #pragma unroll
    for (int j = 0; j < 4; ++j)
        prod[j] = (v8f){1.f, 1.f, 1.f, 1.f, 1.f, 1.f, 1.f, 1.f};

    // Per-r base pointers for this wave (r = wave*4 + j).
    const float* wbr[4];
    #pragma unroll
    for (int j = 0; j < 4; ++j)
        wbr[j] = wb + (size_t)(wave * 4 + j) * (NF * NU);

    for (int f = 0; f < NF; ++f) {
        // Shared A tile for this feature (same for all 4 r's).
        v2f a = *(const v2f*)(xa + (size_t)f * ND);

        #pragma unroll
        for (int j = 0; j < 4; ++j) {
            v2f b;
            b.x = wbr[j][(size_t)f * NU];            // d = 2*half
            b.y = wbr[j][(size_t)f * NU + dplane];   // d = 2*half + 1
            // D = A(16x4) x B(4x16) + 0  -> v_wmma_f32_16x16x4_f32
            v8f t = __builtin_amdgcn_wmma_f32_16x16x4_f32(
                false, a, false, b, (short)0, vzero, false, false);
            prod[j] *= t;   // Hadamard accumulation over features
        }
    }

    // Sum this wave's 4 r-products.
    v8f acc = (prod[0] + prod[1]) + (prod[2] + prod[3]);

    // Cross-wave reduction over the 8 waves (remaining sum over r).
    __shared__ float red[8][32 * 8];
    #pragma unroll
    for (int v = 0; v < 8; ++v)
        red[wave][lane * 8 + v] = acc[v];
    __syncthreads();

    // 256 output elements per block; thread t owns (lane o_l, vgpr-slot v).
    {
        const int o_l = tid & 31;
        const int v   = tid >> 5;
        float s = 0.f;
        #pragma unroll
        for (int w = 0; w < 8; ++w)
            s += red[w][o_l * 8 + v];
        // C/D 16x16 f32 layout: lanes 0-15 -> M=v, lanes 16-31 -> M=v+8; N=o_l&15.
        const int m = v + ((o_l >> 4) << 3);
        const int n = o_l & 15;
        out[(size_t)(b0 + m) * NU + (size_t)(u0 + n)] = s;
    }
}

extern "C" void kernel_launch(void* const* d_in, const int* in_sizes, int n_in,
                              void* d_out, int out_size, void* d_ws, size_t ws_size,
                              hipStream_t stream) {
    const float* X = (const float*)d_in[0];   // [NB, NF, ND] fp32
    const float* W = (const float*)d_in[1];   // [ND, NR, NF, NU] fp32
    float* out = (float*)d_out;               // [NB, NU] fp32

    dim3 grid(NB / 16, NU / 16);              // 64 x 8 blocks
    cp_wmma_kernel<<<grid, dim3(256), 0, stream>>>(X, W, out);
}